// CompositionBlock_17119739642119
// MI455X (gfx1250) — compile-verified
//
#include <hip/hip_runtime.h>
#include <cmath>

typedef float v2f __attribute__((ext_vector_type(2)));
typedef float v8f __attribute__((ext_vector_type(8)));
typedef int   v4i __attribute__((ext_vector_type(4)));

#define N_ROWS  2048     // B*S
#define T_DIM   128
#define C_DIM   128      // output columns of both GEMMs (P and O)
#define KC      32       // K-chunk staged in LDS (divides DX for both stages)
#define LDSS    36       // padded column stride: 144B -> 16B aligned, conflict-free
#define ROWS_PB 32       // rows per block (2 x 16-row tiles)

#if defined(__has_builtin)
# if __has_builtin(__builtin_amdgcn_global_load_async_to_lds_b128)
#  define HAS_ASYNC_LDS 1
# endif
# if __has_builtin(__builtin_amdgcn_s_wait_asynccnt)
#  define WAIT_ASYNC(n) __builtin_amdgcn_s_wait_asynccnt(n)
# endif
#endif
#if defined(HAS_ASYNC_LDS) && !defined(WAIT_ASYNC)
# define WAIT_ASYNC(n) asm volatile("s_wait_asynccnt %0" :: "i"(n))
#endif

typedef __attribute__((address_space(1))) v4i* g_v4i_p;
typedef __attribute__((address_space(3))) v4i* l_v4i_p;

// y[n, c] = tanh( sum_k (tok[n, k/DX] * x[n, k%DX]) * W[c*K + k] + bias[c] )
// DX=64  : x = dep_embeddings, W = W_dep  (K = 8192)
// DX=128 : x = h (stage-1 out), W = W_comp (K = 16384)
template<int DX>
__global__ __launch_bounds__(128)
void gemm_bilin_tanh(const float* __restrict__ tok,
                     const float* __restrict__ x,
                     const float* __restrict__ W,
                     const float* __restrict__ bias,
                     float* __restrict__ y)
{
    constexpr int K   = T_DIM * DX;
    constexpr int NCH = K / KC;
    __shared__ float lds[2][C_DIM * LDSS];

    const int tid  = threadIdx.x;
    const int lane = tid & 31;
    const int wave = tid >> 5;
    const int lrow = lane & 15;   // A row / B col within a 16-tile
    const int kh   = lane >> 4;   // K half: lanes 16..31 hold K=2,3 of each 4-block
    const int rt   = wave & 1;    // row-tile of this wave
    const int ch   = wave >> 1;   // column half (0..1) of this wave
    const int n0   = blockIdx.x * ROWS_PB;
    const int row  = n0 + rt * 16 + lrow;
    const int cbase = ch * 64;

    v8f acc[4] = {};              // 4 x (16x16 f32) tiles = 64 columns

    const float* tokrow = tok + (size_t)row * T_DIM;
    const float* xrow   = x   + (size_t)row * DX;
    const float* gcol   = W   + (size_t)tid * K;   // thread tid stages column c = tid

    // stage chunk c (KC x 128 cols, 16KB) into lds[buf]
    auto issue = [&](int c, int buf) {
        const float* g = gcol + c * KC;
#if defined(HAS_ASYNC_LDS)
        float* l = &lds[buf][tid * LDSS];
        #pragma unroll
        for (int i = 0; i < KC / 4; ++i) {   // 8 async b128 per thread
            __builtin_amdgcn_global_load_async_to_lds_b128(
                (g_v4i_p)(g + i * 4), (l_v4i_p)(l + i * 4), 0, 0);
        }
#else
        #pragma unroll
        for (int i = 0; i < KC / 4; ++i) {
            float4 v = *(const float4*)(g + i * 4);
            *(float4*)&lds[buf][tid * LDSS + i * 4] = v;
        }
        if (c + 1 < NCH) __builtin_prefetch(g + KC, 0, 0);
#endif
    };

    issue(0, 0);
    for (int c = 0; c < NCH; ++c) {
        if (c + 1 < NCH) {
            issue(c + 1, (c + 1) & 1);       // overlap DMA of next chunk
#if defined(HAS_ASYNC_LDS)
            WAIT_ASYNC(KC / 4);              // chunk c landed (in-order), next in flight
        } else {
            WAIT_ASYNC(0);
#endif
        }
        __syncthreads();

        const int   t    = (c * KC) / DX;    // one t per chunk (KC divides DX)
        const int   xoff = (c * KC) % DX;
        const float tokv = tokrow[t];
        const float* B   = &lds[c & 1][0];

        #pragma unroll
        for (int ks = 0; ks < KC / 4; ++ks) {
            const int kl = ks * 4 + kh * 2;
            // A fragment: on-the-fly outer product tok[n,t] * x[n, d..d+1]
            v2f xv = *(const v2f*)(xrow + xoff + kl);
            v2f a; a.x = tokv * xv.x; a.y = tokv * xv.y;
            #pragma unroll
            for (int ct = 0; ct < 4; ++ct) {
                const int col = cbase + ct * 16 + lrow;
                v2f b = *(const v2f*)&B[col * LDSS + kl];
                acc[ct] = __builtin_amdgcn_wmma_f32_16x16x4_f32(
                    false, a, false, b, (short)0, acc[ct], false, false);
            }
        }
        __syncthreads();
    }

    // epilogue: D tile layout -> lane(0-15): M=r, lane(16-31): M=8+r; N=lrow
    const int rbase = rt * 16 + kh * 8;
    #pragma unroll
    for (int ct = 0; ct < 4; ++ct) {
        const int   col = cbase + ct * 16 + lrow;
        const float bc  = bias[col];
        #pragma unroll
        for (int r = 0; r < 8; ++r) {
            y[(size_t)(n0 + rbase + r) * C_DIM + col] = tanhf(acc[ct][r] + bc);
        }
    }
}

// out[b,i,t] = tanh(b_comp[t]) * sum_j Wr[j] + b_red
__global__ __launch_bounds__(256)
void init_out(const float* __restrict__ Wr, const float* __restrict__ b_comp,
              const float* __restrict__ b_red, float* __restrict__ out)
{
    __shared__ float red[256];
    const int tid = threadIdx.x;
    red[tid] = Wr[tid];              // S == 256 == blockDim.x
    __syncthreads();
    for (int s = 128; s > 0; s >>= 1) {
        if (tid < s) red[tid] += red[tid + s];
        __syncthreads();
    }
    const float sumWr = red[0];
    const int idx = blockIdx.x * 256 + tid;
    const int t = idx & (T_DIM - 1);
    out[idx] = tanhf(b_comp[t]) * sumWr + b_red[0];
}

// out[b, head[b,j], t] += Wr[j] * (special[b,j,t] - tanh(b_comp[t]))
__global__ __launch_bounds__(128)
void scatter_heads(const float* __restrict__ special, const int* __restrict__ heads,
                   const float* __restrict__ Wr, const float* __restrict__ b_comp,
                   float* __restrict__ out)
{
    const int n = blockIdx.x;            // n = b*256 + j
    const int head = heads[n];
    if (head >= 256) return;             // root / no-head -> contributes nothing
    const int b = n >> 8;
    const int j = n & 255;
    const int t = threadIdx.x;
    const float w    = Wr[j];
    const float base = tanhf(b_comp[t]);
    const float d    = w * (special[(size_t)n * T_DIM + t] - base);
    atomicAdd(&out[((size_t)(b * 256 + head)) * T_DIM + t], d);
}

extern "C" void kernel_launch(void* const* d_in, const int* in_sizes, int n_in,
                              void* d_out, int out_size, void* d_ws, size_t ws_size,
                              hipStream_t stream)
{
    (void)in_sizes; (void)n_in; (void)out_size; (void)ws_size;
    const float* tok    = (const float*)d_in[0];
    const float* dep    = (const float*)d_in[1];
    const int*   heads  = (const int*)  d_in[2];
    const float* W_dep  = (const float*)d_in[3];
    const float* b_dep  = (const float*)d_in[4];
    const float* W_comp = (const float*)d_in[5];
    const float* b_comp = (const float*)d_in[6];
    const float* W_red  = (const float*)d_in[7];
    const float* b_red  = (const float*)d_in[8];
    float* out = (float*)d_out;

    float* H  = (float*)d_ws;                      // [2048,128] stage-1 activations
    float* SP = H + (size_t)N_ROWS * C_DIM;        // [2048,128] stage-2 "special"

    hipLaunchKernelGGL((gemm_bilin_tanh<64>),  dim3(N_ROWS / ROWS_PB), dim3(128), 0, stream,
                       tok, dep, W_dep, b_dep, H);
    hipLaunchKernelGGL((gemm_bilin_tanh<128>), dim3(N_ROWS / ROWS_PB), dim3(128), 0, stream,
                       tok, H, W_comp, b_comp, SP);
    hipLaunchKernelGGL(init_out,               dim3(1024), dim3(256), 0, stream,
                       W_red, b_comp, b_red, out);
    hipLaunchKernelGGL(scatter_heads,          dim3(2048), dim3(128), 0, stream,
                       SP, heads, W_red, b_comp, out);
}